// GNNActorCritic_30829275251173
// MI455X (gfx1250) — compile-verified
//
#include <hip/hip_runtime.h>
#include <hip/hip_bf16.h>

typedef float v2f __attribute__((ext_vector_type(2)));
typedef float v8f __attribute__((ext_vector_type(8)));

// ---------------------------------------------------------------------------
// Generic f32 WMMA GEMM: C[M,N] = act(A[M,K] @ B[K,N] + bias[N])
// Requirements: M % 16 == 0, N % 16 == 0, K % 4 == 0 (all call sites satisfy).
// One wave computes one 16x16 tile using V_WMMA_F32_16X16X4_F32.
// A 16x4 layout: lanes 0-15 rows M=0..15; VGPR0={K0 | K2}, VGPR1={K1 | K3}
// B 4x16 layout: VGPR0={row K0 lanes0-15 | row K2 lanes16-31}, VGPR1={K1 | K3}
// C/D: VGPR v -> (M=v, N=lane) lanes 0-15 ; (M=v+8, N=lane-16) lanes 16-31
// act: 0=none, 1=relu, 2=tanh
// ---------------------------------------------------------------------------
__global__ void k_gemm_wmma(const float* __restrict__ A,
                            const float* __restrict__ Bm,
                            const float* __restrict__ bias,
                            float* __restrict__ C,
                            int M, int K, int N, int act)
{
    const int lane  = threadIdx.x & 31;
    const int wib   = threadIdx.x >> 5;
    const int wpb   = blockDim.x >> 5;
    const int wave  = blockIdx.x * wpb + wib;
    const int tilesN = N >> 4;
    const int tilesM = M >> 4;
    if (wave >= tilesM * tilesN) return;     // whole-wave exit: EXEC stays full for WMMA waves
    const int tileM = wave / tilesN;
    const int tileN = wave % tilesN;
    const int half  = lane >> 4;             // 0: K even pair, 1: K odd pair base+2
    const int lr    = lane & 15;
    const int m0 = tileM << 4;
    const int n0 = tileN << 4;

    v8f acc = {};
    const float* __restrict__ arow = A + (size_t)(m0 + lr) * K;
    for (int k0 = 0; k0 < K; k0 += 4) {
        const int ka = k0 + 2 * half;
        v2f a; a.x = arow[ka];
               a.y = arow[ka + 1];
        v2f b; b.x = Bm[(size_t)ka       * N + n0 + lr];
               b.y = Bm[(size_t)(ka + 1) * N + n0 + lr];
        acc = __builtin_amdgcn_wmma_f32_16x16x4_f32(
            /*neg_a=*/false, a, /*neg_b=*/false, b,
            /*c_mod=*/(short)0, acc, /*reuse_a=*/false, /*reuse_b=*/false);
    }

    const int col = n0 + lr;
    const float bv = bias ? bias[col] : 0.0f;
#pragma unroll
    for (int v = 0; v < 8; ++v) {
        const int row = m0 + v + 8 * half;
        float val = acc[v] + bv;
        if (act == 1)      val = fmaxf(val, 0.0f);
        else if (act == 2) val = tanhf(val);
        C[(size_t)row * N + col] = val;
    }
}

// ---------------------------------------------------------------------------
// Edge-list construction: int64 edge_index -> int32 src/dst + self loops
// ---------------------------------------------------------------------------
__global__ void k_build_edges(const long long* __restrict__ ei,
                              int* __restrict__ src, int* __restrict__ dst,
                              int E, int Nn)
{
    int i = blockIdx.x * blockDim.x + threadIdx.x;
    if (i >= E + Nn) return;
    if (i < E) { src[i] = (int)ei[i]; dst[i] = (int)ei[(size_t)E + i]; }
    else       { src[i] = i - E;      dst[i] = i - E; }
}

// generic fill
__global__ void k_fill(float* __restrict__ p, float v, int n)
{
    int i = blockIdx.x * blockDim.x + threadIdx.x;
    if (i < n) p[i] = v;
}

// per-(node,head) attention terms: s = <h, a_src>, d = <h, a_dst>
__global__ void k_attn_sd(const float* __restrict__ hlin,
                          const float* __restrict__ as, const float* __restrict__ ad,
                          float* __restrict__ s, float* __restrict__ dv,
                          int Nn, int H, int D)
{
    int i = blockIdx.x * blockDim.x + threadIdx.x;
    if (i >= Nn * H) return;
    const int n = i / H, head = i % H;
    const float* __restrict__ hp = hlin + (size_t)n * H * D + head * D;
    const float* __restrict__ asp = as + head * D;
    const float* __restrict__ adp = ad + head * D;
    float ss = 0.f, sd = 0.f;
    for (int k = 0; k < D; ++k) { float h = hp[k]; ss += h * asp[k]; sd += h * adp[k]; }
    s[i] = ss; dv[i] = sd;
}

__device__ inline void atomicMaxF(float* addr, float val)
{
    unsigned int* ua = reinterpret_cast<unsigned int*>(addr);
    unsigned int old = __float_as_uint(*addr);
    while (__uint_as_float(old) < val) {
        unsigned int assumed = old;
        old = atomicCAS(ua, assumed, __float_as_uint(val));
        if (old == assumed) break;
    }
}

// pass 1 over edges: segment max of leaky_relu(s[src]+d[dst]) per (dst, head)
__global__ void k_edge_max(const int* __restrict__ src, const int* __restrict__ dst,
                           const float* __restrict__ s, const float* __restrict__ dv,
                           float* __restrict__ m, int Eall, int H)
{
    int i = blockIdx.x * blockDim.x + threadIdx.x;
    if (i >= Eall * H) return;
    const int e = i / H, head = i % H;
    float ev = s[src[e] * H + head] + dv[dst[e] * H + head];
    ev = ev > 0.f ? ev : 0.2f * ev;
    atomicMaxF(&m[dst[e] * H + head], ev);
}

// pass 2 over edges: denom[dst,h] += w ; agg[dst,h,:] += w * h[src,h,:]
// (alpha normalization deferred: agg/denom == softmax-weighted sum)
// launched with blockDim = H*D, one block per edge
__global__ void k_edge_accum(const int* __restrict__ src, const int* __restrict__ dst,
                             const float* __restrict__ hlin,
                             const float* __restrict__ s, const float* __restrict__ dv,
                             const float* __restrict__ m,
                             float* __restrict__ denom, float* __restrict__ agg,
                             int Eall, int H, int D)
{
    const int e = blockIdx.x;
    if (e >= Eall) return;
    const int t = threadIdx.x;
    const int head = t / D, dd = t % D;
    const int sN = src[e], dN = dst[e];
    float ev = s[sN * H + head] + dv[dN * H + head];
    ev = ev > 0.f ? ev : 0.2f * ev;
    const float w = expf(ev - m[dN * H + head]);
    if (dd == 0) atomicAdd(&denom[dN * H + head], w);
    atomicAdd(&agg[(size_t)dN * (H * D) + t],
              w * hlin[(size_t)sN * (H * D) + t]);
}

// normalize + bias + relu  (layer-3 has H=1, so mean over heads == identity)
__global__ void k_finalize(const float* __restrict__ agg, const float* __restrict__ denom,
                           const float* __restrict__ bias, float* __restrict__ out,
                           int Nn, int H, int D)
{
    int i = blockIdx.x * blockDim.x + threadIdx.x;
    if (i >= Nn * H * D) return;
    const int n = i / (H * D);
    const int c = i % (H * D);
    const int head = c / D;
    float v = agg[i] / denom[n * H + head] + bias[c];
    out[i] = fmaxf(v, 0.0f);
}

// global mean pool (accumulate)
__global__ void k_pool_accum(const float* __restrict__ f, const long long* __restrict__ batch,
                             float* __restrict__ pooled, float* __restrict__ counts,
                             int Nn, int D)
{
    int i = blockIdx.x * blockDim.x + threadIdx.x;
    if (i >= Nn * D) return;
    const int n = i / D, c = i % D;
    const int b = (int)batch[n];
    atomicAdd(&pooled[b * D + c], f[i]);
    if (c == 0) atomicAdd(&counts[b], 1.0f);
}

__global__ void k_pool_div(float* __restrict__ pooled, const float* __restrict__ counts,
                           int Bn, int D)
{
    int i = blockIdx.x * blockDim.x + threadIdx.x;
    if (i >= Bn * D) return;
    pooled[i] /= fmaxf(counts[i / D], 1.0f);
}

// value head GEMV: value[b] = hC[b,:] @ Wc2[:,0] + bc2
__global__ void k_value(const float* __restrict__ hC, const float* __restrict__ Wc2,
                        const float* __restrict__ bc2, float* __restrict__ value,
                        int Bn, int K)
{
    int b = blockIdx.x * blockDim.x + threadIdx.x;
    if (b >= Bn) return;
    float acc = bc2[0];
    for (int k = 0; k < K; ++k) acc += hC[(size_t)b * K + k] * Wc2[k];
    value[b] = acc;
}

// ---------------------------------------------------------------------------

static inline void launch_gemm(const float* A, const float* B, const float* bias,
                               float* C, int M, int K, int N, int act, hipStream_t s)
{
    const int tiles = (M >> 4) * (N >> 4);
    const int wpb = 4;                       // 4 waves (128 threads) per block
    const int blocks = (tiles + wpb - 1) / wpb;
    k_gemm_wmma<<<blocks, wpb * 32, 0, s>>>(A, B, bias, C, M, K, N, act);
}

extern "C" void kernel_launch(void* const* d_in, const int* in_sizes, int n_in,
                              void* d_out, int out_size, void* d_ws, size_t ws_size,
                              hipStream_t stream)
{
    // ---- inputs (setup_inputs dict order) ----
    const float*      x    = (const float*)d_in[0];
    const long long*  ei   = (const long long*)d_in[1];
    const long long*  batch= (const long long*)d_in[2];
    const float* W1  = (const float*)d_in[3];
    const float* as1 = (const float*)d_in[4];
    const float* ad1 = (const float*)d_in[5];
    const float* b1  = (const float*)d_in[6];
    const float* W2  = (const float*)d_in[7];
    const float* as2 = (const float*)d_in[8];
    const float* ad2 = (const float*)d_in[9];
    const float* b2  = (const float*)d_in[10];
    const float* W3  = (const float*)d_in[11];
    const float* as3 = (const float*)d_in[12];
    const float* ad3 = (const float*)d_in[13];
    const float* b3  = (const float*)d_in[14];
    const float* Wa1 = (const float*)d_in[15];
    const float* ba1 = (const float*)d_in[16];
    const float* Wa2 = (const float*)d_in[17];
    const float* ba2 = (const float*)d_in[18];
    const float* Wc1 = (const float*)d_in[19];
    const float* bc1 = (const float*)d_in[20];
    const float* Wc2 = (const float*)d_in[21];
    const float* bc2 = (const float*)d_in[22];

    // ---- shapes (derived from sizes) ----
    const int Nn   = in_sizes[2];                 // 10000 nodes (multiple of 16)
    const int E    = in_sizes[1] / 2;             // 320000
    const int Eall = E + Nn;                      // + self loops
    const int DIN  = in_sizes[0] / Nn;            // 128
    const int HD   = in_sizes[16];                // hidden dim 64
    const int HEADS= in_sizes[4] / HD;            // 4
    const int HW   = HEADS * HD;                  // 256 (concat width)
    const int NUME = in_sizes[18];                // 1024
    const int Bn   = out_size / (NUME + 1);       // 64 graphs

    // ---- workspace carve-up ----
    char* ws = (char*)d_ws;
    size_t off = 0;
    auto take = [&](size_t bytes) -> char* {
        char* p = ws + off;
        off += (bytes + 255) & ~(size_t)255;
        return p;
    };
    int*   src32 = (int*)  take((size_t)Eall * 4);
    int*   dst32 = (int*)  take((size_t)Eall * 4);
    float* hlin  = (float*)take((size_t)Nn * HW * 4);   // linear output h [N,H*D]
    float* agg   = (float*)take((size_t)Nn * HW * 4);   // unnormalized aggregation
    float* f0    = (float*)take((size_t)Nn * HW * 4);   // layer activations ping
    float* f1    = (float*)take((size_t)Nn * HW * 4);   // layer activations pong
    float* sbuf  = (float*)take((size_t)Nn * HEADS * 4);
    float* dbuf  = (float*)take((size_t)Nn * HEADS * 4);
    float* mbuf  = (float*)take((size_t)Nn * HEADS * 4);
    float* denb  = (float*)take((size_t)Nn * HEADS * 4);
    float* pooled= (float*)take((size_t)Bn * HD * 4);
    float* counts= (float*)take((size_t)Bn * 4);
    float* mlpA  = (float*)take((size_t)Bn * HD * 4);
    float* mlpC  = (float*)take((size_t)Bn * HD * 4);
    (void)ws_size;

    float* actionOut = (float*)d_out;                 // [Bn, NUME]
    float* valueOut  = actionOut + (size_t)Bn * NUME; // [Bn]

    const int T = 256;
    const float NEG_INF = -3.402823466e+38f;

    // ---- edge lists (with self loops) ----
    k_build_edges<<<(Eall + T - 1) / T, T, 0, stream>>>(ei, src32, dst32, E, Nn);

    // ================= GAT layer 1: in=DIN, H heads, concat =================
    launch_gemm(x, W1, nullptr, hlin, Nn, DIN, HW, 0, stream);
    k_attn_sd<<<(Nn * HEADS + T - 1) / T, T, 0, stream>>>(hlin, as1, ad1, sbuf, dbuf, Nn, HEADS, HD);
    k_fill<<<(Nn * HEADS + T - 1) / T, T, 0, stream>>>(mbuf, NEG_INF, Nn * HEADS);
    k_fill<<<(Nn * HEADS + T - 1) / T, T, 0, stream>>>(denb, 0.f, Nn * HEADS);
    k_fill<<<(Nn * HW + T - 1) / T, T, 0, stream>>>(agg, 0.f, Nn * HW);
    k_edge_max<<<(Eall * HEADS + T - 1) / T, T, 0, stream>>>(src32, dst32, sbuf, dbuf, mbuf, Eall, HEADS);
    k_edge_accum<<<Eall, HEADS * HD, 0, stream>>>(src32, dst32, hlin, sbuf, dbuf, mbuf, denb, agg, Eall, HEADS, HD);
    k_finalize<<<(Nn * HW + T - 1) / T, T, 0, stream>>>(agg, denb, b1, f0, Nn, HEADS, HD);

    // ================= GAT layer 2: in=HW, H heads, concat ==================
    launch_gemm(f0, W2, nullptr, hlin, Nn, HW, HW, 0, stream);
    k_attn_sd<<<(Nn * HEADS + T - 1) / T, T, 0, stream>>>(hlin, as2, ad2, sbuf, dbuf, Nn, HEADS, HD);
    k_fill<<<(Nn * HEADS + T - 1) / T, T, 0, stream>>>(mbuf, NEG_INF, Nn * HEADS);
    k_fill<<<(Nn * HEADS + T - 1) / T, T, 0, stream>>>(denb, 0.f, Nn * HEADS);
    k_fill<<<(Nn * HW + T - 1) / T, T, 0, stream>>>(agg, 0.f, Nn * HW);
    k_edge_max<<<(Eall * HEADS + T - 1) / T, T, 0, stream>>>(src32, dst32, sbuf, dbuf, mbuf, Eall, HEADS);
    k_edge_accum<<<Eall, HEADS * HD, 0, stream>>>(src32, dst32, hlin, sbuf, dbuf, mbuf, denb, agg, Eall, HEADS, HD);
    k_finalize<<<(Nn * HW + T - 1) / T, T, 0, stream>>>(agg, denb, b2, f1, Nn, HEADS, HD);

    // ================= GAT layer 3: in=HW, 1 head, mean =====================
    launch_gemm(f1, W3, nullptr, hlin, Nn, HW, HD, 0, stream);
    k_attn_sd<<<(Nn + T - 1) / T, T, 0, stream>>>(hlin, as3, ad3, sbuf, dbuf, Nn, 1, HD);
    k_fill<<<(Nn + T - 1) / T, T, 0, stream>>>(mbuf, NEG_INF, Nn);
    k_fill<<<(Nn + T - 1) / T, T, 0, stream>>>(denb, 0.f, Nn);
    k_fill<<<(Nn * HD + T - 1) / T, T, 0, stream>>>(agg, 0.f, Nn * HD);
    k_edge_max<<<(Eall + T - 1) / T, T, 0, stream>>>(src32, dst32, sbuf, dbuf, mbuf, Eall, 1);
    k_edge_accum<<<Eall, HD, 0, stream>>>(src32, dst32, hlin, sbuf, dbuf, mbuf, denb, agg, Eall, 1, HD);
    k_finalize<<<(Nn * HD + T - 1) / T, T, 0, stream>>>(agg, denb, b3, f0, Nn, 1, HD);

    // ================= global mean pool =====================================
    k_fill<<<(Bn * HD + T - 1) / T, T, 0, stream>>>(pooled, 0.f, Bn * HD);
    k_fill<<<(Bn + T - 1) / T, T, 0, stream>>>(counts, 0.f, Bn);
    k_pool_accum<<<(Nn * HD + T - 1) / T, T, 0, stream>>>(f0, batch, pooled, counts, Nn, HD);
    k_pool_div<<<(Bn * HD + T - 1) / T, T, 0, stream>>>(pooled, counts, Bn, HD);

    // ================= actor / critic heads (WMMA GEMMs) ====================
    launch_gemm(pooled, Wa1, ba1, mlpA, Bn, HD, HD, /*relu*/1, stream);
    launch_gemm(mlpA, Wa2, ba2, actionOut, Bn, HD, NUME, /*tanh*/2, stream);
    launch_gemm(pooled, Wc1, bc1, mlpC, Bn, HD, HD, /*relu*/1, stream);
    k_value<<<1, ((Bn + 63) / 64) * 64, 0, stream>>>(mlpC, Wc2, bc2, valueOut, Bn, HD);
}